// CombinationDecoder_86715389706280
// MI455X (gfx1250) — compile-verified
//
#include <hip/hip_runtime.h>

#define N_TOK 1024
#define DDIM  256
#define MID   256
#define SPAD  260   // row stride in LDS: 16B-aligned for b128, conflict-free columns

typedef __attribute__((ext_vector_type(2))) float v2f;
typedef __attribute__((ext_vector_type(8))) float v8f;

// ---------------------------------------------------------------------------
// fast tanh: hardware v_tanh_f32 on gfx1250 if available, else exp-based
// ---------------------------------------------------------------------------
__device__ __forceinline__ float fast_tanh(float x)
{
#if __has_builtin(__builtin_amdgcn_tanhf)
  return __builtin_amdgcn_tanhf(x);
#else
  float ax = __builtin_fabsf(x);
  float e  = __expf(-2.0f * ax);            // in (0,1], no overflow
  float t  = (1.0f - e) / (1.0f + e);
  return __builtin_copysignf(t, x);
#endif
}

// async global->LDS 16-byte copy (gfx1250 VGLOBAL async op, ASYNCcnt-tracked)
__device__ __forceinline__ void async_copy_b128(const float* gsrc, float* ldst)
{
  unsigned lds_off = (unsigned)(unsigned long long)ldst;  // LDS byte offset
  asm volatile("global_load_async_to_lds_b128 %0, %1, off"
               :: "v"(lds_off), "v"(gsrc) : "memory");
}

__device__ __forceinline__ void wait_asynccnt0()
{
  asm volatile("s_wait_asynccnt 0x0" ::: "memory");
}

// ---------------------------------------------------------------------------
// Kernel B (dominant, ~268M tanh): 16x16 tile of (i,j) pairs per block.
// u already has b1 folded in. u/v tiles staged to LDS with async b128 copies.
// Writes pred = score^T with diagonal and pred[:,0] zeroed, directly to d_out.
// ---------------------------------------------------------------------------
__global__ __launch_bounds__(256)
void score_pred(const float* __restrict__ u, const float* __restrict__ v,
                const float* __restrict__ W2, const float* __restrict__ b2,
                float* __restrict__ pred)
{
  __shared__ float su[16][SPAD];
  __shared__ float sv[16][SPAD];
  __shared__ float sw[MID];

  const int tid = threadIdx.x;
  const int i0  = blockIdx.y * 16;
  const int j0  = blockIdx.x * 16;

  // async-stage 16x256 u tile and 16x256 v tile: 1024 b128 transfers each,
  // 256 threads -> 4 each per tile
  for (int t = tid; t < 16 * (MID / 4); t += 256) {
    int r = t >> 6;             // row 0..15
    int c = (t & 63) * 4;       // col 0..252 step 4
    async_copy_b128(&u[(size_t)(i0 + r) * MID + c], &su[r][c]);
    async_copy_b128(&v[(size_t)(j0 + r) * MID + c], &sv[r][c]);
  }
  sw[tid] = W2[tid];            // blockDim == MID == 256
  wait_asynccnt0();
  __syncthreads();

  const int ti  = tid >> 4;
  const int tj  = tid & 15;
  const int row = i0 + ti;      // score row (i)
  const int col = j0 + tj;      // score col (j)

  float acc = 0.0f;
#pragma unroll 4
  for (int m = 0; m < MID; m += 4) {
    float4 a4 = *(const float4*)&su[ti][m];
    float4 b4 = *(const float4*)&sv[tj][m];
    float4 w4 = *(const float4*)&sw[m];
    acc = __builtin_fmaf(fast_tanh(a4.x + b4.x), w4.x, acc);
    acc = __builtin_fmaf(fast_tanh(a4.y + b4.y), w4.y, acc);
    acc = __builtin_fmaf(fast_tanh(a4.z + b4.z), w4.z, acc);
    acc = __builtin_fmaf(fast_tanh(a4.w + b4.w), w4.w, acc);
  }
  float s = acc + b2[0];

  // pred[col,row] = score[row,col]; zero on diagonal or pred column 0 (row==0)
  pred[(size_t)col * N_TOK + row] = (row == col || row == 0) ? 0.0f : s;
}

// ---------------------------------------------------------------------------
// Kernel A: u = h @ W1[:, :256]^T + b1 ; v = h @ W1[:, 256:]^T  via f32 WMMA.
// One wave per 16x16 output tile; K stepped by 4.
// ---------------------------------------------------------------------------
__global__ __launch_bounds__(32)
void gemm_uv_wmma(const float* __restrict__ h, const float* __restrict__ W1,
                  const float* __restrict__ b1,
                  float* __restrict__ u, float* __restrict__ v)
{
  const int lane = threadIdx.x & 31;
  const int l    = lane & 15;
  const int lh   = lane >> 4;            // which half of the wave
  const int m0   = blockIdx.x * 16;      // MID column tile
  const int i0   = blockIdx.y * 16;      // token row tile
  const int koff = blockIdx.z * DDIM;    // 0 -> u half of W1, 256 -> v half
  float* __restrict__ out = blockIdx.z ? v : u;

  const float* aBase = h  + (size_t)(i0 + l) * DDIM;
  const float* bBase = W1 + (size_t)(m0 + l) * (2 * DDIM) + koff;

  v8f acc = {};
  for (int k0 = 0; k0 < DDIM; k0 += 4) {
    float2 av = *(const float2*)(aBase + k0 + 2 * lh);   // h[i0+l, k0+2lh..+1]
    float2 bv = *(const float2*)(bBase + k0 + 2 * lh);   // W1[m0+l, koff+k0+2lh..+1]
    v2f a = {av.x, av.y};
    v2f b = {bv.x, bv.y};
    acc = __builtin_amdgcn_wmma_f32_16x16x4_f32(false, a, false, b,
                                                (short)0, acc, false, false);
  }
  const float bias = blockIdx.z ? 0.0f : b1[m0 + l];  // fold b1 into u
#pragma unroll
  for (int r = 0; r < 8; ++r)
    out[(size_t)(i0 + r + 8 * lh) * MID + (m0 + l)] = acc[r] + bias;
}

// ---------------------------------------------------------------------------
// Kernel C: per-row log-softmax NLL. score[i,j] = pred[j,i] (j!=i), -inf at j==i.
// Rows 1..N-1 only (pred column-0 zeroing only affects score row 0 -> unused).
// ---------------------------------------------------------------------------
__global__ __launch_bounds__(256)
void row_nll(const float* __restrict__ pred, const int* __restrict__ heads,
             float* __restrict__ rowloss)
{
  __shared__ float red[256];
  const int i   = blockIdx.x + 1;   // row 1..N-1
  const int tid = threadIdx.x;

  float mx = -__builtin_inff();
  for (int j = tid; j < N_TOK; j += 256) {
    float x = (j == i) ? -__builtin_inff() : pred[(size_t)j * N_TOK + i];
    mx = fmaxf(mx, x);
  }
  red[tid] = mx; __syncthreads();
  for (int off = 128; off > 0; off >>= 1) {
    if (tid < off) red[tid] = fmaxf(red[tid], red[tid + off]);
    __syncthreads();
  }
  mx = red[0]; __syncthreads();

  float se = 0.0f;
  for (int j = tid; j < N_TOK; j += 256) {
    float x = (j == i) ? -__builtin_inff() : pred[(size_t)j * N_TOK + i];
    se += __expf(x - mx);
  }
  red[tid] = se; __syncthreads();
  for (int off = 128; off > 0; off >>= 1) {
    if (tid < off) red[tid] += red[tid + off];
    __syncthreads();
  }
  if (tid == 0) {
    int   t  = heads[i];
    float st = (t == i) ? -__builtin_inff() : pred[(size_t)t * N_TOK + i];
    rowloss[i - 1] = (mx + __logf(red[0])) - st;   // -logp[i, t]
  }
}

// ---------------------------------------------------------------------------
// Kernel D: mean of rowloss -> d_out[0] (fresh assignment; graph-replay safe)
// ---------------------------------------------------------------------------
__global__ __launch_bounds__(256)
void reduce_loss(const float* __restrict__ rowloss, float* __restrict__ loss_out)
{
  __shared__ float red[256];
  const int tid = threadIdx.x;
  float s = 0.0f;
  for (int k = tid; k < N_TOK - 1; k += 256) s += rowloss[k];
  red[tid] = s; __syncthreads();
  for (int off = 128; off > 0; off >>= 1) {
    if (tid < off) red[tid] += red[tid + off];
    __syncthreads();
  }
  if (tid == 0) loss_out[0] = red[0] / (float)(N_TOK - 1);
}

extern "C" void kernel_launch(void* const* d_in, const int* in_sizes, int n_in,
                              void* d_out, int out_size, void* d_ws, size_t ws_size,
                              hipStream_t stream)
{
  (void)in_sizes; (void)n_in; (void)out_size; (void)ws_size;

  const float* src   = (const float*)d_in[0];  // (1, N, D) f32
  const int*   heads = (const int*)  d_in[1];  // (N,) int32
  const float* W1    = (const float*)d_in[2];  // (MID, 2D) f32
  const float* b1    = (const float*)d_in[3];  // (MID,)
  const float* W2    = (const float*)d_in[4];  // (1, MID)
  const float* b2    = (const float*)d_in[5];  // (1,)
  float* out = (float*)d_out;                  // [loss, pred(N*N)] f32

  float* u       = (float*)d_ws;                       // N*MID (b1 folded in)
  float* v       = u + (size_t)N_TOK * MID;            // N*MID
  float* rowloss = v + (size_t)N_TOK * MID;            // N-1
  float* pred    = out + 1;

  dim3 gA(MID / 16, N_TOK / 16, 2);
  gemm_uv_wmma<<<gA, dim3(32, 1, 1), 0, stream>>>(src, W1, b1, u, v);

  dim3 gB(N_TOK / 16, N_TOK / 16, 1);
  score_pred<<<gB, dim3(256, 1, 1), 0, stream>>>(u, v, W2, b2, pred);

  row_nll<<<dim3(N_TOK - 1, 1, 1), dim3(256, 1, 1), 0, stream>>>(pred, heads, rowloss);
  reduce_loss<<<dim3(1, 1, 1), dim3(256, 1, 1), 0, stream>>>(rowloss, out);
}